// Correlation_Layer_noNorm_48687749268010
// MI455X (gfx1250) — compile-verified
//
#include <hip/hip_runtime.h>
#include <hip/hip_bf16.h>

// Shapes (hard-coded to the reference): x = (32, 128, 32, 32) f32
//   BN=32, C=128, HW=1024, b=8, n=4  -> 96 GEMMs of (1024x128)x(128x1024)
// Output: 96 * 1024 * 1024 f32 = 402 MB  (store-bandwidth bound: ~17us @ 23.3TB/s)

typedef float v2f __attribute__((ext_vector_type(2)));
typedef float v8f __attribute__((ext_vector_type(8)));
typedef int   v4i __attribute__((ext_vector_type(4)));

#ifndef __has_builtin
#define __has_builtin(x) 0
#endif

#if __has_builtin(__builtin_amdgcn_global_load_async_to_lds_b128) && \
    __has_builtin(__builtin_amdgcn_s_wait_asynccnt)
#define USE_ASYNC_LDS 1
typedef __attribute__((address_space(1))) v4i gv4i_t;   // global int4*
typedef __attribute__((address_space(3))) v4i lv4i_t;   // LDS int4*
#else
#define USE_ASYNC_LDS 0
#endif

#define EPSV 1e-8f

// ---------------- Kernel 1: L2 normalize along channels ----------------
// One thread per (bn, p) spatial position; 1024-strided channel walk, coalesced in p.
__global__ __launch_bounds__(256) void norm_kernel(const float* __restrict__ x,
                                                   float* __restrict__ f) {
    int idx = blockIdx.x * 256 + threadIdx.x;     // 0 .. 32767
    int bn  = idx >> 10;
    int p   = idx & 1023;
    const float* xp = x + (size_t)bn * 131072 + p;   // 131072 = C*HW
    float s = 0.f;
#pragma unroll 8
    for (int c = 0; c < 128; ++c) {
        float v = xp[c << 10];
        s += v * v;
    }
    float inv = 1.0f / (sqrtf(s) + EPSV);
    float* fp = f + (size_t)bn * 131072 + p;
#pragma unroll 8
    for (int c = 0; c < 128; ++c) {
        fp[c << 10] = xp[c << 10] * inv;
    }
}

// ---------------- Kernel 2: pairwise correlation GEMMs via WMMA ----------------
// block = 256 threads (8 wave32). Each block -> one 128x128 output tile of one pair.
// Wave (4x2 grid) computes a 32x64 sub-tile = 2x4 WMMA 16x16 accumulators.
__global__ __launch_bounds__(256) void corr_kernel(const float* __restrict__ f,
                                                   float* __restrict__ out) {
    __shared__ float As[128 * 128];   // A-slab: [c][p]  (64 KB)
    __shared__ float Bs[128 * 128];   // B-slab: [c][q]  (64 KB)

    int blk = blockIdx.x;     // 0 .. 6143
    int g   = blk >> 6;       // pair id 0..95  (= b*12 + i*3 + k)
    int t   = blk & 63;       // tile id in 8x8 grid
    int pt  = t >> 3, qt = t & 7;

    int b  = g / 12;
    int r  = g % 12;
    int i  = r / 3;
    int kk = r % 3;
    int j  = kk + (kk >= i ? 1 : 0);   // jj[i][kk]

    const float* A = f + (size_t)(b * 4 + j) * 131072;  // rows p come from view j
    const float* B = f + (size_t)(b * 4 + i) * 131072;  // cols q come from view i
    int p0 = pt << 7, q0 = qt << 7;

    int tid = threadIdx.x;

    // ---- stage 128x128 slabs into LDS (4096 float4 each) ----
    for (int v = tid; v < 4096; v += 256) {
        int c    = v >> 5;             // channel row
        int col4 = (v & 31) << 2;      // column (floats), 16B aligned
#if USE_ASYNC_LDS
        __builtin_amdgcn_global_load_async_to_lds_b128(
            (gv4i_t*)(A + (size_t)c * 1024 + p0 + col4),
            (lv4i_t*)(&As[(c << 7) + col4]), 0, 0);
        __builtin_amdgcn_global_load_async_to_lds_b128(
            (gv4i_t*)(B + (size_t)c * 1024 + q0 + col4),
            (lv4i_t*)(&Bs[(c << 7) + col4]), 0, 0);
#else
        float4 va = *(const float4*)(A + (size_t)c * 1024 + p0 + col4);
        *(float4*)(&As[(c << 7) + col4]) = va;
        float4 vb = *(const float4*)(B + (size_t)c * 1024 + q0 + col4);
        *(float4*)(&Bs[(c << 7) + col4]) = vb;
#endif
    }
#if USE_ASYNC_LDS
    __builtin_amdgcn_s_wait_asynccnt(0);
#endif
    __syncthreads();

    int wave = tid >> 5;
    int lane = tid & 31;
    int lo = lane & 15, hi = lane >> 4;
    int mw = (wave >> 1) << 5;   // wave row base: 0,32,64,96
    int nw = (wave & 1) << 6;    // wave col base: 0,64

    v8f zero = {0.f, 0.f, 0.f, 0.f, 0.f, 0.f, 0.f, 0.f};
    v8f acc[2][4];
    for (int rr = 0; rr < 2; ++rr)
        for (int cc = 0; cc < 4; ++cc)
            acc[rr][cc] = zero;

    // K slot mapping (same for A and B frags): K = kb + v + 2*hi
    for (int kb = 0; kb < 128; kb += 4) {
        int k0 = kb + (hi << 1);
        int rowK0 = k0 << 7;
        int rowK1 = (k0 + 1) << 7;

        v2f a0, a1;
        a0.x = As[rowK0 + mw + lo];       a0.y = As[rowK1 + mw + lo];
        a1.x = As[rowK0 + mw + 16 + lo];  a1.y = As[rowK1 + mw + 16 + lo];

        v2f b0, b1, b2, b3;
        b0.x = Bs[rowK0 + nw + lo];       b0.y = Bs[rowK1 + nw + lo];
        b1.x = Bs[rowK0 + nw + 16 + lo];  b1.y = Bs[rowK1 + nw + 16 + lo];
        b2.x = Bs[rowK0 + nw + 32 + lo];  b2.y = Bs[rowK1 + nw + 32 + lo];
        b3.x = Bs[rowK0 + nw + 48 + lo];  b3.y = Bs[rowK1 + nw + 48 + lo];

        acc[0][0] = __builtin_amdgcn_wmma_f32_16x16x4_f32(false, a0, false, b0, (short)0, acc[0][0], false, false);
        acc[0][1] = __builtin_amdgcn_wmma_f32_16x16x4_f32(false, a0, false, b1, (short)0, acc[0][1], false, false);
        acc[0][2] = __builtin_amdgcn_wmma_f32_16x16x4_f32(false, a0, false, b2, (short)0, acc[0][2], false, false);
        acc[0][3] = __builtin_amdgcn_wmma_f32_16x16x4_f32(false, a0, false, b3, (short)0, acc[0][3], false, false);
        acc[1][0] = __builtin_amdgcn_wmma_f32_16x16x4_f32(false, a1, false, b0, (short)0, acc[1][0], false, false);
        acc[1][1] = __builtin_amdgcn_wmma_f32_16x16x4_f32(false, a1, false, b1, (short)0, acc[1][1], false, false);
        acc[1][2] = __builtin_amdgcn_wmma_f32_16x16x4_f32(false, a1, false, b2, (short)0, acc[1][2], false, false);
        acc[1][3] = __builtin_amdgcn_wmma_f32_16x16x4_f32(false, a1, false, b3, (short)0, acc[1][3], false, false);
    }

    // ---- store: D layout = VGPR d, lane l -> (M = d + 8*(l>>4), N = l&15) ----
    float* outg = out + (size_t)g * 1024 * 1024;
    for (int rr = 0; rr < 2; ++rr) {
        for (int cc = 0; cc < 4; ++cc) {
            int Mbase = p0 + mw + rr * 16 + (hi << 3);
            int N     = q0 + nw + cc * 16 + lo;
#pragma unroll
            for (int d = 0; d < 8; ++d) {
                outg[(size_t)(Mbase + d) * 1024 + N] = acc[rr][cc][d];
            }
        }
    }
}

extern "C" void kernel_launch(void* const* d_in, const int* in_sizes, int n_in,
                              void* d_out, int out_size, void* d_ws, size_t ws_size,
                              hipStream_t stream) {
    const float* x = (const float*)d_in[0];
    float* f   = (float*)d_ws;          // 32*128*1024*4 = 16 MB normalized features
    float* out = (float*)d_out;         // 96*1024*1024 f32

    norm_kernel<<<128, 256, 0, stream>>>(x, f);
    corr_kernel<<<6144, 256, 0, stream>>>(f, out);
}